// Kernel_63196148793917
// MI455X (gfx1250) — compile-verified
//
#include <hip/hip_runtime.h>
#include <hip/hip_bf16.h>

typedef __attribute__((ext_vector_type(2))) float v2f;
typedef __attribute__((ext_vector_type(8))) float v8f;

#define ZTOT   100000
#define KDIM   1024
#define NDIM   1024      // DO*DI = 32*32
#define HDIM   128
#define NPATHS 6
#define BLOCK_Z 32       // z rows per block (2 WMMA M-tiles)
#define NWAVES  8
#define NT      8        // 16-wide N tiles per wave (8*16 = 128 cols/wave, 8 waves = 1024)
#define MT      2        // 16-tall M tiles per block

// SH constants
#define C0  0.28209479177387814f
#define C1  0.4886025119029199f
#define C2A 1.0925484305920792f
#define C2B 0.31539156525252005f
#define C2C 0.5462742152960396f

__global__ __launch_bounds__(256) void sh_tensor_fused(
    const float* __restrict__ r,         // (Z,3)
    const float* __restrict__ W1,        // (1,H)
    const float* __restrict__ b1,        // (H)
    const float* __restrict__ W2,        // (H,6)
    const float* __restrict__ b2,        // (6)
    const float* __restrict__ cg,        // (K,32,32)
    const float* __restrict__ ylm_mix,   // (K,9)
    const float* __restrict__ rf_mix,    // (K,6)
    const float* __restrict__ norm_coef, // (32,32,2)
    float* __restrict__ out)             // (Z,32,32)
{
    extern __shared__ float smem[];
    float* zdata = smem;                  // [32][16]: Y[0..8], R[0..5], selFlag
    float* S     = smem + BLOCK_Z * 16;   // [K/4][32][4] = 32768 floats (128 KB)

    const int tid   = threadIdx.x;
    const int lane  = tid & 31;
    const int wave  = tid >> 5;
    const int zbase = blockIdx.x * BLOCK_Z;

    // ---------------- Phase A: per-row SH basis + radial MLP (32 lanes) ----------
    if (tid < BLOCK_Z) {
        const int z = zbase + tid;
        const float x  = r[z * 3 + 0];
        const float y  = r[z * 3 + 1];
        const float zc = r[z * 3 + 2];
        const float r2      = x * x + y * y + zc * zc;
        const float safe_r2 = (r2 > 0.f) ? r2 : 1.f;
        const float inv_r   = 1.f / sqrtf(safe_r2);
        const float inv_r2  = 1.f / safe_r2;
        const float radii   = sqrtf(r2);

        float* zd = &zdata[tid * 16];
        zd[0] = C0;
        zd[1] = C1 * y  * inv_r;
        zd[2] = C1 * zc * inv_r;
        zd[3] = C1 * x  * inv_r;
        zd[4] = C2A * x * y  * inv_r2;
        zd[5] = C2A * y * zc * inv_r2;
        zd[6] = C2B * (2.f * zc * zc - x * x - y * y) * inv_r2;
        zd[7] = C2A * x * zc * inv_r2;
        zd[8] = C2C * (x * x - y * y) * inv_r2;

        float R[NPATHS];
        #pragma unroll
        for (int p = 0; p < NPATHS; ++p) R[p] = b2[p];
        for (int j = 0; j < HDIM; ++j) {
            float h = fmaf(radii, W1[j], b1[j]);
            h = fmaxf(h, 0.f);
            #pragma unroll
            for (int p = 0; p < NPATHS; ++p)
                R[p] = fmaf(h, W2[j * NPATHS + p], R[p]);
        }
        #pragma unroll
        for (int p = 0; p < NPATHS; ++p) zd[9 + p] = R[p];
        zd[15] = (r2 == 0.f) ? 1.f : 0.f;   // norm_coef selector
    }
    __syncthreads();

    // ---------------- Phase B: S[z,k] = (rf_mix.R)*(ylm_mix.Y) into LDS ----------
    {
        const int zloc = lane;             // each thread owns one z-row
        float Yz[9], Rz[NPATHS];
        #pragma unroll
        for (int q = 0; q < 9; ++q)      Yz[q] = zdata[zloc * 16 + q];
        #pragma unroll
        for (int p = 0; p < NPATHS; ++p) Rz[p] = zdata[zloc * 16 + 9 + p];

        // warp-uniform k: broadcast rf_mix/ylm_mix loads
        for (int k = wave; k < KDIM; k += NWAVES) {
            float rk = 0.f, yk = 0.f;
            #pragma unroll
            for (int p = 0; p < NPATHS; ++p) rk = fmaf(rf_mix[k * NPATHS + p], Rz[p], rk);
            #pragma unroll
            for (int q = 0; q < 9; ++q)      yk = fmaf(ylm_mix[k * 9 + q], Yz[q], yk);
            // layout [k>>2][z][k&3] -> A-fragment = one aligned 8B LDS read
            S[(k >> 2) * (BLOCK_Z * 4) + zloc * 4 + (k & 3)] = rk * yk;
        }
    }
    __syncthreads();

    // ---------------- Phase C: C[32 x 1024] += S[32 x 1024] * cg[1024 x 1024] ----
    // V_WMMA_F32_16X16X4_F32 fragments (wave32):
    //   A: lane m = lane&15 holds row M=m; VGPR v holds K = v + (lane>=16 ? 2 : 0)
    //   B: lane holds col N = lane&15;     VGPR v holds K = v + (lane>=16 ? 2 : 0)
    //   C/D: VGPR rr holds row M = rr + (lane>=16 ? 8 : 0), col N = lane&15
    v8f acc[MT][NT];
    #pragma unroll
    for (int mt = 0; mt < MT; ++mt)
        #pragma unroll
        for (int nt = 0; nt < NT; ++nt)
            acc[mt][nt] = (v8f){0.f, 0.f, 0.f, 0.f, 0.f, 0.f, 0.f, 0.f};

    const int m     = lane & 15;
    const int koff  = (lane >> 4) << 1;          // 0 or 2
    const int nbase = wave * (NT * 16) + (lane & 15);

    for (int k0 = 0; k0 < KDIM; k0 += 4) {
        v2f a[MT];
        #pragma unroll
        for (int mt = 0; mt < MT; ++mt)
            a[mt] = *(const v2f*)&S[(k0 >> 2) * (BLOCK_Z * 4) + (mt * 16 + m) * 4 + koff];

        const float* cgk = cg + (size_t)(k0 + koff) * NDIM;
        #pragma unroll
        for (int nt = 0; nt < NT; ++nt) {
            const int n = nbase + nt * 16;
            v2f b;
            b.x = cgk[n];            // K = k0+koff
            b.y = cgk[NDIM + n];     // K = k0+koff+1
            #pragma unroll
            for (int mt = 0; mt < MT; ++mt)
                acc[mt][nt] = __builtin_amdgcn_wmma_f32_16x16x4_f32(
                    false, a[mt], false, b, (short)0, acc[mt][nt], false, false);
        }
    }

    // ---------------- Epilogue: apply norm_coef[.., radii==0] and store ----------
    const int rowoff = (lane >> 4) << 3;         // 0 or 8
    #pragma unroll
    for (int nt = 0; nt < NT; ++nt) {
        const int n = nbase + nt * 16;
        const float nc0 = norm_coef[n * 2 + 0];
        const float nc1 = norm_coef[n * 2 + 1];
        #pragma unroll
        for (int mt = 0; mt < MT; ++mt) {
            #pragma unroll
            for (int rr = 0; rr < 8; ++rr) {
                const int zloc = mt * 16 + rowoff + rr;
                const float ncv = (zdata[zloc * 16 + 15] > 0.5f) ? nc1 : nc0;
                out[(size_t)(zbase + zloc) * NDIM + n] = acc[mt][nt][rr] * ncv;
            }
        }
    }
}

extern "C" void kernel_launch(void* const* d_in, const int* in_sizes, int n_in,
                              void* d_out, int out_size, void* d_ws, size_t ws_size,
                              hipStream_t stream) {
    (void)in_sizes; (void)n_in; (void)d_ws; (void)ws_size; (void)out_size;
    const float* r         = (const float*)d_in[0];
    const float* W1        = (const float*)d_in[1];
    const float* b1        = (const float*)d_in[2];
    const float* W2        = (const float*)d_in[3];
    const float* b2        = (const float*)d_in[4];
    const float* cg        = (const float*)d_in[5];
    const float* ylm_mix   = (const float*)d_in[6];
    const float* rf_mix    = (const float*)d_in[7];
    const float* norm_coef = (const float*)d_in[8];
    float* out             = (float*)d_out;

    const int nblocks = (ZTOT + BLOCK_Z - 1) / BLOCK_Z;            // 3125
    const size_t lds_bytes = (BLOCK_Z * 16 + (KDIM / 4) * BLOCK_Z * 4) * sizeof(float); // 130 KB
    sh_tensor_fused<<<nblocks, 256, lds_bytes, stream>>>(
        r, W1, b1, W2, b2, cg, ylm_mix, rf_mix, norm_coef, out);
}